// LoFloatMXFakeQuant_40149354283229
// MI455X (gfx1250) — compile-verified
//
#include <hip/hip_runtime.h>
#include <cstdint>

// ---------------------------------------------------------------------------
// MX (block-32) fake-quant to P3109 binary8p4 elements with binary8p4 scales.
//
// Roofline: 8192^2 fp32 in + out = 512 MB streamed once. At 23.3 TB/s HBM the
// floor is ~22 us; ~12 VALU ops/element is far below the VALU ceiling, so the
// kernel is purely bandwidth-bound. Strategy:
//   * CDNA5 async copy engine: global_load_async_to_lds_b128 with an 8-deep
//     per-wave pipeline (s_wait_asynccnt), decoupling HBM latency from the
//     wave without spending VGPRs on in-flight data.
//   * Wave-uniform loop control forced into SALU via readfirstlane; steady
//     state has constant wait immediates and no pipeline bookkeeping.
//   * B128 accesses; non-temporal hints both directions (512 MB stream cannot
//     be retained in the 192 MB L2).
//   * One MX block (32 floats) = 8 consecutive lanes' float4s; amax reduction
//     is 3 intra-group shuffle-xor ops.
// ---------------------------------------------------------------------------

typedef float vf4 __attribute__((ext_vector_type(4)));

#define DEPTH 8u           // async pipeline depth (stages per wave)
#define WPB   8u           // waves per 256-thread block
#define CHUNK_BYTES 512u   // one wave-chunk: 32 lanes x 16B
#define RING_BYTES (DEPTH * CHUNK_BYTES)   // 4 KB per-wave LDS ring

// floor(log2(|x|)) for normal x via exponent bits. Denormal/zero return -127,
// which is below every clamp threshold we use, matching the reference
// (sub-emin exponents clamp to emin; sub -10 scale exponents -> 0).
__device__ __forceinline__ int floorlog2(float af) {
    return (int)((__float_as_uint(af) >> 23) & 0xFFu) - 127;
}
// Exact 2^e for e in [-126,127] (dead-path wraparound is neutralized by the
// scale==0 zeroing below).
__device__ __forceinline__ float exp2i(int e) {
    return __uint_as_float((uint32_t)(e + 127) << 23);
}

// RNE quantize y to the binary8p4 grid (m=3, emin=-7, emax=7, saturate +-240).
__device__ __forceinline__ float q_b8p4(float y) {
    int ey = floorlog2(__builtin_fabsf(y));
    ey = ey < -7 ? -7 : (ey > 7 ? 7 : ey);
    // y * 2^(3-ey) is exact (pow2 scaling); rintf == RNE == jnp.round
    float q = rintf(y * exp2i(3 - ey)) * exp2i(ey - 3);
    return fminf(fmaxf(q, -240.0f), 240.0f);
}

// Full MX fake-quant of one float4 per lane; block = 8-lane group.
__device__ __forceinline__ vf4 mxq4(vf4 v) {
    float a = fmaxf(fmaxf(__builtin_fabsf(v.x), __builtin_fabsf(v.y)),
                    fmaxf(__builtin_fabsf(v.z), __builtin_fabsf(v.w)));
    // amax over the 8-lane group (masks 1/2/4 stay inside the group)
    a = fmaxf(a, __shfl_xor(a, 1, 32));
    a = fmaxf(a, __shfl_xor(a, 2, 32));
    a = fmaxf(a, __shfl_xor(a, 4, 32));

    int se = floorlog2(a) - 7;             // shared exponent
    float scale, inv;
    if (se > 7) {                          // scale clips to max-normal 240
        scale = 240.0f; inv = 1.0f / 240.0f;
    } else {                               // exact power-of-two scale
        scale = exp2i(se); inv = exp2i(-se);
    }
    if (se < -10) scale = 0.0f;            // scale underflows the b8p4 grid
    // (in that path inv may be garbage/Inf; q clamps to finite +-240 or NaN
    //  is discarded by v_min/v_max, and *0 reproduces the reference's 0.)

    vf4 r;
    r.x = q_b8p4(v.x * inv) * scale;
    r.y = q_b8p4(v.y * inv) * scale;
    r.z = q_b8p4(v.z * inv) * scale;
    r.w = q_b8p4(v.w * inv) * scale;
    return r;
}

__device__ __forceinline__ void async_issue(uint32_t ldsOff, const char* gp) {
    asm volatile("global_load_async_to_lds_b128 %0, %1, off th:TH_LOAD_NT"
                 :: "v"(ldsOff), "v"(gp) : "memory");
}
__device__ __forceinline__ vf4 lds_read128(uint32_t ldsOff) {
    vf4 v;
    asm volatile("ds_load_b128 %0, %1" : "=v"(v) : "v"(ldsOff));
    asm volatile("s_wait_dscnt 0x0" ::: "memory");
    return v;
}

// ---------------------------------------------------------------------------
// Main streaming kernel: 8-deep async global->LDS pipeline per wave.
// Structure: prologue (fill) -> steady state (wait 7 / consume / reissue) ->
// drain (wait 0 / consume the last <=8). All loop control is wave-uniform
// scalar (readfirstlane), so guards become s_cmp/s_cbranch, not EXEC masks.
// ---------------------------------------------------------------------------
__global__ __launch_bounds__(256) void
mx_fq_main(const char* __restrict__ xbytes, vf4* __restrict__ o4,
           unsigned int chunks, unsigned int totalWaves) {
    __shared__ char lds_raw[WPB * RING_BYTES];   // 32 KB / block

    const unsigned int tid  = threadIdx.x;
    const unsigned int lane = tid & 31u;
    const unsigned int wv   = tid >> 5;
    // wave-uniform global wave id, forced into an SGPR
    const unsigned int gw =
        __builtin_amdgcn_readfirstlane(blockIdx.x * WPB + wv);
    if (gw >= chunks) return;

    // Flat LDS addresses carry the wave-relative byte offset in the low 32
    // bits; hardware adds LDS_BASE for both the async writer and ds_load.
    const uint32_t ldsLane =
        (uint32_t)(uintptr_t)&lds_raw[0] + wv * RING_BYTES + lane * 16u;

    const unsigned int nIter = (chunks - 1u - gw) / totalWaves + 1u;
    const unsigned int P     = nIter < DEPTH ? nIter : DEPTH;

    const uint64_t stride = (uint64_t)totalWaves * CHUNK_BYTES;  // bytes
    const char* gp = xbytes + (uint64_t)gw * CHUNK_BYTES + lane * 16u;
    vf4*        op = o4 + (uint64_t)gw * 32u + lane;
    const uint64_t ostride = (uint64_t)totalWaves * 32u;         // vf4 elems

    // ---- prologue: fill the pipeline --------------------------------------
    for (unsigned int s = 0; s < P; ++s) {
        async_issue(ldsLane + s * CHUNK_BYTES, gp);
        gp += stride;
    }

    uint32_t stageOff = 0;

    // ---- steady state: pipeline stays exactly DEPTH deep ------------------
    for (unsigned int i = DEPTH; i < nIter; ++i) {
        asm volatile("s_wait_asynccnt 0x7" ::: "memory");  // oldest complete
        const uint32_t slot = ldsLane + stageOff;
        vf4 v = lds_read128(slot);                 // includes s_wait_dscnt 0
        async_issue(slot, gp);                     // reuse the freed slot
        gp += stride;
        __builtin_nontemporal_store(mxq4(v), op);
        op += ostride;
        stageOff = (stageOff + CHUNK_BYTES) & (RING_BYTES - 1u);
    }

    // ---- drain: everything left is in flight; wait once, then consume -----
    asm volatile("s_wait_asynccnt 0x0" ::: "memory");
    for (unsigned int s = 0; s < P; ++s) {
        vf4 v = lds_read128(ldsLane + stageOff);
        __builtin_nontemporal_store(mxq4(v), op);
        op += ostride;
        stageOff = (stageOff + CHUNK_BYTES) & (RING_BYTES - 1u);
    }
}

// ---------------------------------------------------------------------------
// Tail kernel: up to 3 leftover 32-element blocks (n % 128, n % 32 == 0).
// ---------------------------------------------------------------------------
__global__ __launch_bounds__(32) void
mx_fq_tail(const vf4* __restrict__ x4, vf4* __restrict__ o4,
           unsigned int startElem, unsigned int nBlocks32) {
    const unsigned int lane = threadIdx.x & 31u;
    const bool act = lane < nBlocks32 * 8u;
    vf4 v = (vf4)0.0f;
    const unsigned int idx4 = startElem / 4u + lane;
    if (act) v = x4[idx4];
    vf4 r = mxq4(v);     // all 32 lanes execute (shuffles need full EXEC)
    if (act) o4[idx4] = r;
}

extern "C" void kernel_launch(void* const* d_in, const int* in_sizes, int n_in,
                              void* d_out, int out_size, void* d_ws, size_t ws_size,
                              hipStream_t stream) {
    (void)n_in; (void)out_size; (void)d_ws; (void)ws_size;
    const char* xb = (const char*)d_in[0];
    vf4* o4 = (vf4*)d_out;

    const long long n = in_sizes[0];                 // 8192*8192 = 64Mi floats
    if (n <= 0) return;
    const unsigned int chunks = (unsigned int)(n / 128);     // 512B wave-chunks
    const unsigned int tailBlocks =
        (unsigned int)((n - (long long)chunks * 128) / 32);

    if (chunks) {
        unsigned int blocks = (chunks + WPB - 1u) / WPB;
        if (blocks > 2048u) blocks = 2048u;          // 16K waves, grid-stride
        const unsigned int totalWaves = blocks * WPB;
        mx_fq_main<<<blocks, 256, 0, stream>>>(xb, o4, chunks, totalWaves);
    }
    if (tailBlocks) {
        mx_fq_tail<<<1, 32, 0, stream>>>((const vf4*)xb, o4,
                                         chunks * 128u, tailBlocks);
    }
}